// GridPooling_223338299640
// MI455X (gfx1250) — compile-verified
//
#include <hip/hip_runtime.h>
#include <math.h>

// ---------- sizes ----------
#define BATCH 2048     // S*N
#define HDIM  64
#define ADIM  1024
#define G2DIM 64
#define PPTS  4096
#define KPAD  80       // 69 padded up to a multiple of BK (16)

// GEMM tiling
#define BM 64
#define BN 128
#define BK 16

typedef __attribute__((ext_vector_type(2))) float v2f;
typedef __attribute__((ext_vector_type(8))) float v8f;

// =====================================================================
// Tiled fp32 WMMA GEMM:  D = A(MxK) * B(KxN) (+bias0 +bias1), row-major.
// 256 threads / 8 waves; block tile 64x128; K-step 16; double-buffered LDS.
// Wave w: M sub-tile (w&3)*16, N strip (w>>2)*64 -> 4 accumulators;
// A-fragment registers reused across 4 WMMAs.
// Pipeline per iteration: [global loads for tile it+1] -> [16 WMMA on tile it]
// -> [LDS stores for it+1] -> barrier.  Prefetch tile it+2 via
// __builtin_prefetch (global_prefetch_b8).
// WMMA f32 16x16x4 operand layout (ISA 7.12.2):
//   A: lanes 0-15 hold K={k,k+1} of row M=lane; lanes 16-31 hold K={k+2,k+3}
//   C/D: VGPR r -> M=r (lanes 0-15) / M=r+8 (lanes 16-31), N=lane&15
// =====================================================================
__global__ __launch_bounds__(256)
void wmma_gemm_tiled(const float* __restrict__ A, int lda,
                     const float* __restrict__ B, int ldb,
                     const float* __restrict__ bias0,
                     const float* __restrict__ bias1,
                     float* __restrict__ D, int ldd,
                     int K)
{
    __shared__ float As[2][BM][BK + 1];
    __shared__ float Bs[2][BK][BN];

    const int tid  = threadIdx.x;
    const int wave = tid >> 5;
    const int lane = tid & 31;
    const int half = lane >> 4;
    const int lm   = lane & 15;

    const int m0 = blockIdx.y * BM;
    const int n0 = blockIdx.x * BN;

    const int wm    = wave & 3;          // 0..3 -> M sub-tile
    const int nbase = (wave >> 2) * 64;  // 0 or 64 -> N strip
    const int mloc  = wm * 16 + lm;

    // Staging indices (coalesced): A: 4 lanes per 64B row segment;
    // B: 32 lanes per 128-float row, 2 rounds.
    const int arow = tid >> 2;           // 0..63
    const int akk  = (tid & 3) * 4;      // 0,4,8,12
    const int brow0 = tid >> 5;          // q=0: rows 0..7
    const int bcol  = (tid & 31) * 4;    // 0..124

    const float* Aptr = A + (size_t)(m0 + arow) * lda + akk;
    const float* Bptr0 = B + (size_t)brow0 * ldb + n0 + bcol;
    const float* Bptr1 = B + (size_t)(brow0 + 8) * ldb + n0 + bcol;

    v8f acc[4] = {};

    const int nIter = K / BK;

    // ---- prologue: stage tile 0 ----
    {
        const float4 av = *(const float4*)Aptr;
        As[0][arow][akk + 0] = av.x;
        As[0][arow][akk + 1] = av.y;
        As[0][arow][akk + 2] = av.z;
        As[0][arow][akk + 3] = av.w;
        const float4 bv0 = *(const float4*)Bptr0;
        const float4 bv1 = *(const float4*)Bptr1;
        *(float4*)&Bs[0][brow0][bcol]     = bv0;
        *(float4*)&Bs[0][brow0 + 8][bcol] = bv1;
    }
    __syncthreads();

    for (int it = 0; it < nIter; ++it) {
        const int cur = it & 1;
        const bool has_next = (it + 1) < nIter;

        // ---- issue global loads for tile it+1 (fly under the WMMAs) ----
        float4 av_n, bv0_n, bv1_n;
        if (has_next) {
            const int kn = (it + 1) * BK;
            av_n  = *(const float4*)(Aptr + kn);
            bv0_n = *(const float4*)(Bptr0 + (size_t)kn * ldb);
            bv1_n = *(const float4*)(Bptr1 + (size_t)kn * ldb);
        }
        // ---- prefetch tile it+2 into the cache hierarchy ----
        if (it + 2 < nIter) {
            const int kp = (it + 2) * BK;
            __builtin_prefetch(Aptr + kp, 0, 3);
            __builtin_prefetch(Bptr0 + (size_t)kp * ldb, 0, 3);
            __builtin_prefetch(Bptr1 + (size_t)kp * ldb, 0, 3);
        }

        // ---- compute: 4 K-steps x 4 N-tiles = 16 WMMA, A-fragment reused ----
#pragma unroll
        for (int ks = 0; ks < BK; ks += 4) {
            const int ka = ks + 2 * half;
            v2f a;
            a.x = As[cur][mloc][ka];
            a.y = As[cur][mloc][ka + 1];
#pragma unroll
            for (int nt = 0; nt < 4; ++nt) {
                const int nc = nbase + nt * 16 + lm;
                v2f b;
                b.x = Bs[cur][ka][nc];
                b.y = Bs[cur][ka + 1][nc];
                acc[nt] = __builtin_amdgcn_wmma_f32_16x16x4_f32(
                              false, a, false, b, (short)0, acc[nt], false, false);
            }
        }

        // ---- commit tile it+1 into the other buffer ----
        if (has_next) {
            const int nxt = cur ^ 1;
            As[nxt][arow][akk + 0] = av_n.x;
            As[nxt][arow][akk + 1] = av_n.y;
            As[nxt][arow][akk + 2] = av_n.z;
            As[nxt][arow][akk + 3] = av_n.w;
            *(float4*)&Bs[nxt][brow0][bcol]     = bv0_n;
            *(float4*)&Bs[nxt][brow0 + 8][bcol] = bv1_n;
        }
        __syncthreads();
    }

    // ---- store with fused bias ----
#pragma unroll
    for (int nt = 0; nt < 4; ++nt) {
        const int col = n0 + nbase + nt * 16 + lm;
        float badd = 0.f;
        if (bias0) badd += bias0[col];
        if (bias1) badd += bias1[col];
#pragma unroll
        for (int r = 0; r < 8; ++r) {
            const int row = m0 + wm * 16 + r + 8 * half;
            D[(size_t)row * ldd + col] = acc[nt][r] + badd;
        }
    }
}

// =====================================================================
// Occupancy: one block per pedestrian; LDS-atomic 64-bin histogram.
// =====================================================================
__global__ __launch_bounds__(256)
void occupancy_kernel(const float* __restrict__ end_pos,
                      const float* __restrict__ scene,
                      float* __restrict__ occ)
{
    __shared__ int hist[G2DIM];
    const int b = blockIdx.x;
    const int s = b >> 6;                 // 64 peds per sequence

    for (int i = threadIdx.x; i < G2DIM; i += blockDim.x) hist[i] = 0;
    __syncthreads();

    const float ex  = end_pos[2 * b];
    const float ey  = end_pos[2 * b + 1];
    const float tlx = ex - 1.0f, tly = ey + 1.0f;
    const float brx = ex + 1.0f, bry = ey - 1.0f;

    const float* sp = scene + (size_t)s * PPTS * 2;
    for (int j = threadIdx.x; j < PPTS; j += blockDim.x) {
        const float sx = sp[2 * j];
        const float sy = sp[2 * j + 1];
        const bool oob = (sx >= brx) | (sx <= tlx) | (sy >= tly) | (sy <= bry);
        if (!oob) {
            const int cx = (int)floorf((sx - tlx) * 4.0f);   // G/NS = 4
            const int cy = (int)floorf((tly - sy) * 4.0f);
            const int cell = cx + cy * 8;
            if (cell >= 0 && cell < G2DIM) atomicAdd(&hist[cell], 1);
        }
    }
    __syncthreads();
    for (int g = threadIdx.x; g < G2DIM; g += blockDim.x)
        occ[(size_t)b * G2DIM + g] = (float)hist[g];
}

// =====================================================================
// Fused attention: e[b,g] = sum_a relu(occ*W_enc[a] + base[b,a]) * w_full[a]
// then softmax over g and ctx[b] = sum_g alpha*occ.  One block per b.
// (ReLU sits between the outer product and the contraction -> not a GEMM;
//  att (B,G2,A) = 536MB is never materialized.)
// =====================================================================
__global__ __launch_bounds__(256)
void attention_kernel(const float* __restrict__ occ,
                      const float* __restrict__ base,
                      const float* __restrict__ W_enc,
                      const float* __restrict__ w_full,
                      const float* __restrict__ b_full,
                      float* __restrict__ ctx)
{
    __shared__ float wenc_s[ADIM];
    __shared__ float wf_s[ADIM];
    __shared__ float base_s[ADIM];
    __shared__ float occ_s[G2DIM];
    __shared__ float e_s[G2DIM];

    const int b = blockIdx.x;
    for (int i = threadIdx.x; i < ADIM; i += blockDim.x) {
        wenc_s[i] = W_enc[i];
        wf_s[i]   = w_full[i];
        base_s[i] = base[(size_t)b * ADIM + i];
    }
    for (int i = threadIdx.x; i < G2DIM; i += blockDim.x)
        occ_s[i] = occ[(size_t)b * G2DIM + i];
    __syncthreads();

    const int wave = threadIdx.x >> 5;
    const int lane = threadIdx.x & 31;
#pragma unroll
    for (int gi = 0; gi < 8; ++gi) {
        const int g = wave * 8 + gi;
        const float o = occ_s[g];
        float p = 0.f;
        for (int a = lane; a < ADIM; a += 32) {
            const float t = fmaf(o, wenc_s[a], base_s[a]);
            p += fmaxf(t, 0.f) * wf_s[a];
        }
        for (int off = 16; off; off >>= 1) p += __shfl_xor(p, off, 32);
        if (lane == 0) e_s[g] = p + b_full[0];
    }
    __syncthreads();

    if (threadIdx.x == 0) {
        float m = -3.402823466e38f;
        for (int g = 0; g < G2DIM; ++g) m = fmaxf(m, e_s[g]);
        float se = 0.f, cx = 0.f;
        for (int g = 0; g < G2DIM; ++g) {
            const float ee = expf(e_s[g] - m);
            se += ee;
            cx += ee * occ_s[g];
        }
        ctx[b] = cx / se;
    }
}

// =====================================================================
// Pack Xp (B x 80) = [ctx | h | end_pos | rel_pos | 0-pad]
// =====================================================================
__global__ void build_xp(const float* __restrict__ ctx,
                         const float* __restrict__ h,
                         const float* __restrict__ end_pos,
                         const float* __restrict__ rel_pos,
                         float* __restrict__ Xp)
{
    const int idx = blockIdx.x * blockDim.x + threadIdx.x;
    if (idx >= BATCH * KPAD) return;
    const int b = idx / KPAD, k = idx % KPAD;
    float v;
    if (k == 0)       v = ctx[b];
    else if (k < 65)  v = h[(size_t)b * HDIM + (k - 1)];
    else if (k < 67)  v = end_pos[b * 2 + (k - 65)];
    else if (k < 69)  v = rel_pos[b * 2 + (k - 67)];
    else              v = 0.f;
    Xp[idx] = v;
}

// Pad W_out (69x1024) to Wp (80x1024), zero-filled rows 69..79
__global__ void build_wp(const float* __restrict__ W_out, float* __restrict__ Wp)
{
    const int idx = blockIdx.x * blockDim.x + threadIdx.x;
    if (idx >= KPAD * ADIM) return;
    const int k = idx / ADIM;
    Wp[idx] = (k < 69) ? W_out[idx] : 0.f;
}

// =====================================================================
// Batch-axis layernorm (mean/var over rows) in two passes + ReLU
// =====================================================================
__global__ void zero_stats(float* __restrict__ sums, float* __restrict__ ssq)
{
    const int i = blockIdx.x * blockDim.x + threadIdx.x;
    if (i < ADIM) { sums[i] = 0.f; ssq[i] = 0.f; }
}

__global__ __launch_bounds__(256)
void col_stats(const float* __restrict__ y,
               float* __restrict__ sums, float* __restrict__ ssq)
{
    const int r0 = blockIdx.x * 128;   // 16 blocks * 128 rows = 2048
    const int t  = threadIdx.x;
    float s[4] = {0, 0, 0, 0}, q[4] = {0, 0, 0, 0};
    for (int r = 0; r < 128; ++r) {
        const float* row = y + (size_t)(r0 + r) * ADIM;
#pragma unroll
        for (int c = 0; c < 4; ++c) {
            const float v = row[t + 256 * c];
            s[c] += v;
            q[c] += v * v;
        }
    }
#pragma unroll
    for (int c = 0; c < 4; ++c) {
        atomicAdd(&sums[t + 256 * c], s[c]);
        atomicAdd(&ssq[t + 256 * c], q[c]);
    }
}

__global__ void normalize_kernel(float* __restrict__ y,
                                 const float* __restrict__ sums,
                                 const float* __restrict__ ssq,
                                 const float* __restrict__ gamma,
                                 const float* __restrict__ beta)
{
    const int idx = blockIdx.x * blockDim.x + threadIdx.x;
    if (idx >= BATCH * ADIM) return;
    const int a = idx & (ADIM - 1);
    const float inv = 1.0f / (float)BATCH;
    const float mu  = sums[a] * inv;
    const float var = ssq[a] * inv - mu * mu;
    const float v = gamma[a] * (y[idx] - mu) * rsqrtf(var + 1e-5f) + beta[a];
    y[idx] = fmaxf(v, 0.f);
}

// =====================================================================
extern "C" void kernel_launch(void* const* d_in, const int* in_sizes, int n_in,
                              void* d_out, int out_size, void* d_ws, size_t ws_size,
                              hipStream_t stream)
{
    const float* h_states = (const float*)d_in[0];   // (B, 64)
    const float* end_pos  = (const float*)d_in[1];   // (B, 2)
    const float* rel_pos  = (const float*)d_in[2];   // (B, 2)
    const float* scene    = (const float*)d_in[3];   // (32, 4096, 2)
    const float* W_enc    = (const float*)d_in[4];   // (1, 1024)
    const float* b_enc    = (const float*)d_in[5];   // (1024,)
    const float* W_dec    = (const float*)d_in[6];   // (64, 1024)
    const float* b_dec    = (const float*)d_in[7];   // (1024,)
    const float* w_full   = (const float*)d_in[8];   // (1024, 1)
    const float* b_full   = (const float*)d_in[9];   // (1,)
    const float* W_out    = (const float*)d_in[10];  // (69, 1024)
    const float* b_out    = (const float*)d_in[11];  // (1024,)
    const float* W_mlp    = (const float*)d_in[12];  // (1024, 1024)
    const float* b_mlp    = (const float*)d_in[13];  // (1024,)
    const float* gamma    = (const float*)d_in[14];  // (1024,)
    const float* beta     = (const float*)d_in[15];  // (1024,)
    // d_in[16] seq_start_end: uniform 64-ped sequences by construction

    float* y = (float*)d_out;                        // (B, 1024)

    // workspace layout (floats)
    float* ws   = (float*)d_ws;
    float* occ  = ws;                                //  131072
    float* base = occ  + BATCH * G2DIM;              // 2097152
    float* ctxv = base + BATCH * ADIM;               //    2048
    float* Xp   = ctxv + BATCH;                      //  163840
    float* Wp   = Xp   + BATCH * KPAD;               //   81920
    float* pool = Wp   + KPAD * ADIM;                // 2097152
    float* sums = pool + BATCH * ADIM;               //    1024
    float* ssq  = sums + ADIM;                       //    1024

    const dim3 blk256(256);
    const dim3 gemm_grid(ADIM / BN, BATCH / BM);     // (8, 32)

    // 1) occupancy histograms
    occupancy_kernel<<<dim3(BATCH), blk256, 0, stream>>>(end_pos, scene, occ);

    // 2) base = h @ W_dec + b_dec + b_enc   (WMMA f32, M=2048 N=1024 K=64)
    wmma_gemm_tiled<<<gemm_grid, blk256, 0, stream>>>(
        h_states, HDIM, W_dec, ADIM, b_dec, b_enc, base, ADIM, HDIM);

    // 3) fused attention -> ctx
    attention_kernel<<<dim3(BATCH), blk256, 0, stream>>>(
        occ, base, W_enc, w_full, b_full, ctxv);

    // 4) pack padded pool inputs
    build_xp<<<dim3((BATCH * KPAD + 255) / 256), blk256, 0, stream>>>(
        ctxv, h_states, end_pos, rel_pos, Xp);
    build_wp<<<dim3((KPAD * ADIM + 255) / 256), blk256, 0, stream>>>(W_out, Wp);

    // 5) pool = Xp @ Wp + b_out   (WMMA f32, K=80)
    wmma_gemm_tiled<<<gemm_grid, blk256, 0, stream>>>(
        Xp, KPAD, Wp, ADIM, b_out, nullptr, pool, ADIM, KPAD);

    // 6) y = pool @ W_mlp + b_mlp (WMMA f32, K=1024) -> d_out
    wmma_gemm_tiled<<<gemm_grid, blk256, 0, stream>>>(
        pool, ADIM, W_mlp, ADIM, b_mlp, nullptr, y, ADIM, ADIM);

    // 7) batch-axis layernorm + relu, in place on d_out
    zero_stats<<<dim3(4), blk256, 0, stream>>>(sums, ssq);
    col_stats<<<dim3(BATCH / 128), blk256, 0, stream>>>(y, sums, ssq);
    normalize_kernel<<<dim3((BATCH * ADIM) / 256), blk256, 0, stream>>>(
        y, sums, ssq, gamma, beta);
}